// KLDLoss_70609262346548
// MI455X (gfx1250) — compile-verified
//
#include <hip/hip_runtime.h>
#include <math.h>

#define CLAMP_EPS 1e-7f

typedef __attribute__((ext_vector_type(2))) float v2f;
typedef __attribute__((ext_vector_type(8))) float v8f;

struct F4 { float x, y, z, w; };

// ---------------------------------------------------------------------------
// Per-row loss: KLD distance between two rotated 2D Gaussians, mapped through
// 1 - 1/(tau + log1p(sqrt(dist))) and scaled by weight.  ALPHA=TAU=1.
// Uses native transcendentals (v_sin/v_cos/v_log): theta is in [0,1) so the
// libm argument-reduction slow paths are dead code we don't want to carry.
// ---------------------------------------------------------------------------
__device__ __forceinline__ float kld_loss_row(float p0, float p1, float p2, float p3, float p4,
                                              float t0, float t1, float t2, float t3, float t4,
                                              float w) {
  // Sigma terms, pred
  float wp = fmaxf(p2, CLAMP_EPS), hp = fmaxf(p3, CLAMP_EPS);
  float w2p = 0.25f * wp * wp, h2p = 0.25f * hp * hp;
  float cp = __cosf(p4), sp = __sinf(p4);
  float ap = w2p * cp * cp + h2p * sp * sp;
  float bp = (w2p - h2p) * sp * cp;
  float dp = w2p * sp * sp + h2p * cp * cp;
  float det_p = ap * dp - bp * bp;

  // Sigma terms, target
  float wt = fmaxf(t2, CLAMP_EPS), ht = fmaxf(t3, CLAMP_EPS);
  float w2t = 0.25f * wt * wt, h2t = 0.25f * ht * ht;
  float ct = __cosf(t4), st = __sinf(t4);
  float at = w2t * ct * ct + h2t * st * st;
  float bt = (w2t - h2t) * st * ct;
  float dt_ = w2t * st * st + h2t * ct * ct;
  float det_t = at * dt_ - bt * bt;

  float dx = p0 - t0, dy = p1 - t1;
  float inv_detp = 1.0f / det_p;
  float xy = 0.5f * (dx * dx * dp - 2.0f * dx * dy * bp + dy * dy * ap) * inv_detp;
  float whr = 0.5f * (dp * at - 2.0f * bp * bt + ap * dt_) * inv_detp
            + 0.5f * (__logf(det_p) - __logf(det_t)) - 1.0f;
  float dist = xy + whr;             // ALPHA == 1
  dist = sqrtf(fmaxf(dist, 0.0f));
  float loss = 1.0f - 1.0f / (1.0f + log1pf(dist));   // TAU == 1
  return loss * w;
}

// ---------------------------------------------------------------------------
// Wave32 reduction via V_WMMA_F32_16X16X4_F32.
//   A[m][0] = x_m (lanes 0-15, VGPR0), A[m][2] = x_{m+16} (lanes 16-31, VGPR0),
//   A[m][1] = A[m][3] = 0, B = ones(4x16)
//   => D[m][n] = x_m + x_{m+16} for all n.
// Lane L in [0,15] holds D[0..7][L] in its 8 C VGPRs; lanes 16-31 hold rows
// 8..15.  Summing the 8 VGPRs gives per-half partial sums; one xor-16 shuffle
// merges them.  Requires EXEC == all-ones (callers must not early-return).
// ---------------------------------------------------------------------------
__device__ __forceinline__ float wave_reduce_wmma(float x) {
  v2f a; a[0] = x;    a[1] = 0.0f;
  v2f b; b[0] = 1.0f; b[1] = 1.0f;
  v8f c = {};
  c = __builtin_amdgcn_wmma_f32_16x16x4_f32(
      /*neg_a=*/false, a, /*neg_b=*/false, b,
      /*c_mod=*/(short)0, c, /*reuse_a=*/false, /*reuse_b=*/false);
  float h = ((c[0] + c[1]) + (c[2] + c[3])) + ((c[4] + c[5]) + (c[6] + c[7]));
  h += __shfl_xor(h, 16, 32);
  return h;
}

// ---------------------------------------------------------------------------
// Main kernel: each thread processes 4 rows (five b128 loads per tensor + one
// b128 of weights), accumulates loss in f32, reduces wave->block->f64 atomic.
// Tail rows (N % 4) are handled in the finalize kernel to keep this hot
// kernel free of a duplicated scalar codepath.
// ---------------------------------------------------------------------------
__global__ void kld_main(const float* __restrict__ pred,
                         const float* __restrict__ target,
                         const float* __restrict__ weight,
                         double* __restrict__ acc_ws,
                         int n_groups) {
  const int gid = blockIdx.x * blockDim.x + threadIdx.x;
  float acc = 0.0f;

  if (gid < n_groups) {
    const F4* p4 = (const F4*)(pred   + (size_t)gid * 20);
    const F4* t4 = (const F4*)(target + (size_t)gid * 20);
    F4 pa = p4[0], pb = p4[1], pc = p4[2], pd = p4[3], pe = p4[4];
    F4 ta = t4[0], tb = t4[1], tc = t4[2], td = t4[3], te = t4[4];
    F4 wv = *(const F4*)(weight + (size_t)gid * 4);

    acc += kld_loss_row(pa.x, pa.y, pa.z, pa.w, pb.x,
                        ta.x, ta.y, ta.z, ta.w, tb.x, wv.x);
    acc += kld_loss_row(pb.y, pb.z, pb.w, pc.x, pc.y,
                        tb.y, tb.z, tb.w, tc.x, tc.y, wv.y);
    acc += kld_loss_row(pc.z, pc.w, pd.x, pd.y, pd.z,
                        tc.z, tc.w, td.x, td.y, td.z, wv.z);
    acc += kld_loss_row(pd.w, pe.x, pe.y, pe.z, pe.w,
                        td.w, te.x, te.y, te.z, te.w, wv.w);
  }

  // EXEC is all-ones here (guard above only skipped work, no early return).
  float wsum = wave_reduce_wmma(acc);

  __shared__ float s[32];
  const int lane = threadIdx.x & 31;
  const int wid  = threadIdx.x >> 5;
  if (lane == 0) s[wid] = wsum;
  __syncthreads();
  if (threadIdx.x == 0) {
    float bsum = 0.0f;
    const int nw = (blockDim.x + 31) >> 5;
    for (int i = 0; i < nw; ++i) bsum += s[i];
    atomicAdd(acc_ws, (double)bsum);   // global_atomic_add_f64
  }
}

__global__ void kld_zero(double* acc) {
  if (threadIdx.x == 0 && blockIdx.x == 0) acc[0] = 0.0;
}

// Finalize: fold in the <=3 tail rows, then write mean.  Single thread; runs
// after kld_main on the same stream.
__global__ void kld_finalize(const float* __restrict__ pred,
                             const float* __restrict__ target,
                             const float* __restrict__ weight,
                             const double* __restrict__ acc,
                             float* __restrict__ out, int n_rem, int N) {
  if (threadIdx.x == 0 && blockIdx.x == 0) {
    double total = acc[0];
    for (int r = 0; r < n_rem; ++r) {
      int row = N - n_rem + r;
      const float* p = pred + (size_t)row * 5;
      const float* t = target + (size_t)row * 5;
      total += (double)kld_loss_row(p[0], p[1], p[2], p[3], p[4],
                                    t[0], t[1], t[2], t[3], t[4], weight[row]);
    }
    out[0] = (float)(total / (double)N);   // LOSS_WEIGHT == 1
  }
}

// ---------------------------------------------------------------------------
extern "C" void kernel_launch(void* const* d_in, const int* in_sizes, int n_in,
                              void* d_out, int out_size, void* d_ws, size_t ws_size,
                              hipStream_t stream) {
  const float* pred   = (const float*)d_in[0];
  const float* target = (const float*)d_in[1];
  const float* weight = (const float*)d_in[2];
  float* out = (float*)d_out;
  double* acc = (double*)d_ws;

  const int N = in_sizes[0] / 5;       // rows
  const int groups = N / 4;            // 4 rows per thread
  const int rem = N % 4;

  kld_zero<<<1, 1, 0, stream>>>(acc);

  const int block = 256;               // 8 waves per block on wave32
  int grid = (groups + block - 1) / block;
  if (grid < 1) grid = 1;
  kld_main<<<grid, block, 0, stream>>>(pred, target, weight, acc, groups);

  kld_finalize<<<1, 1, 0, stream>>>(pred, target, weight, acc, out, rem, N);
}